// CenterLoss_28819230556441
// MI455X (gfx1250) — compile-verified
//
#include <hip/hip_runtime.h>

// ---------------------------------------------------------------------------
// CenterLoss on MI455X (gfx1250, wave32).
// mean_i clamp( ||x_i - centers[labels_i]||^2 , 1e-12, 1e12 )
//
// Bandwidth-bound (~16.8 MB @ 23.3 TB/s => ~0.7us floor). Strategy:
//  * GLOBAL_LOAD_ASYNC_TO_LDS_B128 (ASYNCcnt) to stream x rows + gathered
//    center rows directly into LDS (no VGPR round-trip), double-buffered:
//    issue chunk k+1's 8 copies -> s_wait_asynccnt 0x8 -> compute chunk k.
//  * per-wave 16-sample tile reduced with V_WMMA_F32_16X16X4_F32:
//      A = (x-c)^2  (documented A layout: lane L, vgpr v -> M=L&15, K=2*(L>>4)+v)
//      B = all-ones (layout-invariant) => every C[M][N] = sum_k d^2[M,k]
//  * deterministic two-stage reduction (no float atomics -> bitwise stable)
// ---------------------------------------------------------------------------

typedef __attribute__((ext_vector_type(2))) float v2f;
typedef __attribute__((ext_vector_type(8))) float v8f;

#define FEAT_D   512
#define KC       32          // K-chunk staged in LDS per pipeline stage
#define KP       36          // padded LDS row stride (dwords): 16B rows, conflict-free banks
#define NCH      (FEAT_D / KC)   // 16 chunks
#define CLAMP_LO 1e-12f
#define CLAMP_HI 1e12f

__global__ __launch_bounds__(32)
void center_dist_wmma_kernel(const float* __restrict__ x,
                             const float* __restrict__ centers,
                             const long long* __restrict__ labels,
                             float* __restrict__ partials)
{
    __shared__ float ldsX[2][16 * KP];   // double-buffered x tile
    __shared__ float ldsC[2][16 * KP];   // double-buffered gathered-center tile

    const int lane = threadIdx.x;          // 0..31 (wave32)
    const int s0   = blockIdx.x * 16;      // this wave's 16 samples

    // labels for rows 0..15 (values < 50000 -> low 32 bits suffice)
    int lab32 = 0;
    if (lane < 16) lab32 = (int)labels[s0 + lane];

    // ---- per-lane async-copy geometry: lane copies 16B of row (4*it + lane/8)
    const int rsub = lane >> 3;            // 0..3
    const int csub = (lane & 7) * 4;       // dword col within chunk, %4==0
    unsigned long long xb[4], cb[4];       // global byte addresses at kc=0
    unsigned lX0[4], lX1[4], lC0[4], lC1[4];   // LDS byte offsets (per buffer)
    #pragma unroll
    for (int it = 0; it < 4; ++it) {
        const int row    = it * 4 + rsub;
        const int labRow = __shfl(lab32, row, 32);
        xb[it] = (unsigned long long)(const void*)(x       + (size_t)(s0 + row) * FEAT_D + csub);
        cb[it] = (unsigned long long)(const void*)(centers + (size_t)labRow    * FEAT_D + csub);
        const int t = row * KP + csub;     // dword offset in tile (16B aligned)
        lX0[it] = (unsigned)(uintptr_t)(const void*)&ldsX[0][t];
        lX1[it] = (unsigned)(uintptr_t)(const void*)&ldsX[1][t];
        lC0[it] = (unsigned)(uintptr_t)(const void*)&ldsC[0][t];
        lC1[it] = (unsigned)(uintptr_t)(const void*)&ldsC[1][t];
    }

    // issue one chunk's 8 async b128 copies (2048B x-tile + 2048B c-tile)
    #define ISSUE_CHUNK(LXA, LCA, KCBYTES)                                       \
        do {                                                                     \
            _Pragma("unroll")                                                    \
            for (int it = 0; it < 4; ++it) {                                     \
                asm volatile("global_load_async_to_lds_b128 %0, %1, off"         \
                             :: "v"(LXA[it]), "v"(xb[it] + (KCBYTES)) : "memory");\
                asm volatile("global_load_async_to_lds_b128 %0, %1, off"         \
                             :: "v"(LCA[it]), "v"(cb[it] + (KCBYTES)) : "memory");\
            }                                                                    \
        } while (0)

    const int mrow = lane & 15;            // A-matrix row owned by this lane
    const int koff = (lane >> 4) << 1;     // K offset within 4-chunk: 0 or 2

    v8f acc = {};                          // C/D accumulator (16x16 f32)
    const v2f ones = {1.0f, 1.0f};         // B = all-ones (layout-invariant)

    auto compute = [&](const float* __restrict__ LX, const float* __restrict__ LC) {
        #pragma unroll
        for (int k0 = 0; k0 < KC; k0 += 4) {
            const int o = mrow * KP + k0 + koff;    // even dword -> 8B aligned
            const float2 xv = *(const float2*)(LX + o);
            const float2 cv = *(const float2*)(LC + o);
            const float dx = xv.x - cv.x;
            const float dy = xv.y - cv.y;
            v2f a; a.x = dx * dx; a.y = dy * dy;    // A = squared diffs
            // 8-arg form: (neg_a, A, neg_b, B, c_mod, C, reuse_a, reuse_b)
            acc = __builtin_amdgcn_wmma_f32_16x16x4_f32(
                false, a, false, ones, (short)0, acc, false, false);
        }
        // drain LDS reads before the async engine may overwrite this buffer
        asm volatile("s_wait_dscnt 0x0" ::: "memory");
    };

    // ---- software pipeline: prefetch chunk 0, then (prefetch k+1 | compute k)
    ISSUE_CHUNK(lX0, lC0, 0ull);
    for (int c = 0; c < NCH; c += 2) {
        if (c + 1 < NCH) {
            ISSUE_CHUNK(lX1, lC1, (unsigned long long)(c + 1) * (KC * 4));
            asm volatile("s_wait_asynccnt 0x8" ::: "memory");  // chunk c landed
        } else {
            asm volatile("s_wait_asynccnt 0x0" ::: "memory");
        }
        compute(ldsX[0], ldsC[0]);
        if (c + 2 < NCH) {
            ISSUE_CHUNK(lX0, lC0, (unsigned long long)(c + 2) * (KC * 4));
            asm volatile("s_wait_asynccnt 0x8" ::: "memory");  // chunk c+1 landed
        } else {
            asm volatile("s_wait_asynccnt 0x0" ::: "memory");
        }
        compute(ldsX[1], ldsC[1]);
    }
    #undef ISSUE_CHUNK

    // C layout: lane L, vgpr v holds C[M = v + 8*(L>>4)][N = L&15]; with B=ones
    // all N columns identical => acc[v] = dist[v + 8*(L>>4)].
    // Pick each sample once: lanes 0..7 -> samples 0..7, lanes 16..23 -> 8..15.
    float myv = 0.0f;
    const bool activeA = (lane < 8);
    const bool activeB = (lane >= 16) && (lane < 24);
    #pragma unroll
    for (int v = 0; v < 8; ++v) {
        const float t = acc[v];
        if ((activeA && lane == v) || (activeB && (lane - 16) == v)) myv = t;
    }
    float val = (activeA || activeB)
                    ? fminf(fmaxf(myv, CLAMP_LO), CLAMP_HI)
                    : 0.0f;

    // deterministic wave butterfly sum (fixed order)
    #pragma unroll
    for (int w = 16; w >= 1; w >>= 1)
        val += __shfl_xor(val, w, 32);

    if (lane == 0) partials[blockIdx.x] = val;
}

__global__ __launch_bounds__(32)
void center_reduce_kernel(const float* __restrict__ partials,
                          float* __restrict__ out,
                          int n, float invB)
{
    const int lane = threadIdx.x;
    float s = 0.0f;
    for (int i = lane; i < n; i += 32)     // fixed order -> deterministic
        s += partials[i];
    #pragma unroll
    for (int w = 16; w >= 1; w >>= 1)
        s += __shfl_xor(s, w, 32);
    if (lane == 0) out[0] = s * invB;
}

extern "C" void kernel_launch(void* const* d_in, const int* in_sizes, int n_in,
                              void* d_out, int out_size, void* d_ws, size_t ws_size,
                              hipStream_t stream)
{
    const float*     x       = (const float*)d_in[0];      // (4096, 512) f32
    const float*     centers = (const float*)d_in[1];      // (50000, 512) f32
    const long long* labels  = (const long long*)d_in[2];  // (4096,) i64
    float*           out     = (float*)d_out;              // scalar f32
    float*           part    = (float*)d_ws;               // 256 partials

    const int batch = in_sizes[2];          // 4096
    const int nblk  = batch / 16;           // one wave per 16 samples -> 256 blocks

    center_dist_wmma_kernel<<<nblk, 32, 0, stream>>>(x, centers, labels, part);
    center_reduce_kernel<<<1, 32, 0, stream>>>(part, out, nblk, 1.0f / (float)batch);
}